// GraphNet_5643587027542
// MI455X (gfx1250) — compile-verified
//
#include <hip/hip_runtime.h>
#include <math.h>

typedef __attribute__((ext_vector_type(2))) float v2f;
typedef __attribute__((ext_vector_type(8))) float v8f;

#define WPB 8   // waves per block in edge kernel (256 threads)

// ---------------------------------------------------------------------------
// Utility: zero 32-bit words (works for float 0.0f and int 0)
// ---------------------------------------------------------------------------
__global__ void zero_words(unsigned* __restrict__ p, long n) {
    long i = (long)blockIdx.x * blockDim.x + threadIdx.x;
    if (i < n) p[i] = 0u;
}

// out-degree of each node (same for all 3 layers)
__global__ void deg_kernel(long E, const int* __restrict__ row, int* __restrict__ deg) {
    long e = (long)blockIdx.x * blockDim.x + threadIdx.x;
    if (e < E) atomicAdd(&deg[row[e]], 1);
}

// per-graph node count and per-graph edge count (= sum of deg over nodes in graph)
__global__ void gcnt_kernel(long N, const int* __restrict__ batch, const int* __restrict__ deg,
                            float* __restrict__ g_n_cnt, float* __restrict__ g_e_cnt) {
    long n = (long)blockIdx.x * blockDim.x + threadIdx.x;
    if (n < N) {
        int g = batch[n];
        atomicAdd(&g_n_cnt[g], 1.0f);
        atomicAdd(&g_e_cnt[g], (float)deg[n]);
    }
}

// ---------------------------------------------------------------------------
// Edge MLP via f32 WMMA. One wave handles a 16-edge tile.
//   feat(e) = [x[row] | x[col] | ein[e] | u[batch[row]]]   (widths 1/1/1/1 or 16/16/16/16)
//   ea = W2^T . relu(W1^T . feat + b1) + b2
// Writes ea to eout and atomically accumulates ea into node_accum keyed by row.
// ---------------------------------------------------------------------------
template<bool L1>
__launch_bounds__(32 * WPB)
__global__ void edge_kernel(long E,
                            const int* __restrict__ row, const int* __restrict__ col,
                            const int* __restrict__ batch,
                            const float* __restrict__ x, const float* __restrict__ ein,
                            const float* __restrict__ u,
                            const float* __restrict__ W1, const float* __restrict__ B1,
                            const float* __restrict__ W2, const float* __restrict__ B2,
                            float* __restrict__ eout, float* __restrict__ node_accum) {
    __shared__ float lds[WPB][16 * 17];   // per-wave 16x16 transpose buffer (padded)

    const int lane = threadIdx.x & 31;
    const int wave = threadIdx.x >> 5;
    const int m    = lane & 15;   // A: edge row / B,C,D: column index
    const int kh   = lane >> 4;   // K half-select (K%4 in {0,1} vs {2,3})

    const long tile = (long)blockIdx.x * WPB + wave;
    const long base = tile * 16;
    long e = base + m;
    const long eS = (e < E) ? e : (E - 1);   // clamp (keeps EXEC full for WMMA)

    const int r = row[eS];
    const int c = col[eS];
    const int g = batch[r];

    // C init = broadcast b1 over the tile's hidden column j = m
    const float b1m = B1[m];
    v8f acc;
#pragma unroll
    for (int v = 0; v < 8; ++v) acc[v] = b1m;

    if constexpr (L1) {
        // din = 4 : [x[r], x[c], ew[e], u[g]] ; single K=4 WMMA
        v2f a;
        if (kh == 0) { a.x = x[r];    a.y = x[c]; }
        else         { a.x = ein[eS]; a.y = u[g]; }
        const int kb = 2 * kh;
        v2f b; b.x = W1[kb * 16 + m]; b.y = W1[(kb + 1) * 16 + m];
        acc = __builtin_amdgcn_wmma_f32_16x16x4_f32(false, a, false, b, (short)0, acc, false, false);
    } else {
        // din = 64 : four 16-wide segments, 16 chained K=4 WMMAs
        const float* p0 = x   + (size_t)r  * 16;
        const float* p1 = x   + (size_t)c  * 16;
        const float* p2 = ein + (size_t)eS * 16;
        const float* p3 = u   + (size_t)g  * 16;
        v2f a[16];
#pragma unroll
        for (int t = 0; t < 4; ++t) {
            const int off = 4 * t + 2 * kh;
            a[t]      = *(const v2f*)(p0 + off);
            a[4 + t]  = *(const v2f*)(p1 + off);
            a[8 + t]  = *(const v2f*)(p2 + off);
            a[12 + t] = *(const v2f*)(p3 + off);
        }
#pragma unroll
        for (int t = 0; t < 16; ++t) {
            const int kb = 4 * t + 2 * kh;
            v2f b; b.x = W1[kb * 16 + m]; b.y = W1[(kb + 1) * 16 + m];
            acc = __builtin_amdgcn_wmma_f32_16x16x4_f32(false, a[t], false, b, (short)0, acc, false, false);
        }
    }

    // ReLU, then transpose h (D layout -> A layout) through LDS
    float* Lp = lds[wave];
#pragma unroll
    for (int v = 0; v < 8; ++v) {
        const float hv = fmaxf(acc[v], 0.0f);
        Lp[(v + 8 * kh) * 17 + m] = hv;      // [edge_row][hidden_k]
    }
    __syncthreads();

    const float b2m = B2[m];
    v8f acc2;
#pragma unroll
    for (int v = 0; v < 8; ++v) acc2[v] = b2m;
#pragma unroll
    for (int t = 0; t < 4; ++t) {
        const int kb = 4 * t + 2 * kh;
        v2f a2; a2.x = Lp[m * 17 + kb]; a2.y = Lp[m * 17 + kb + 1];
        v2f b;  b.x  = W2[kb * 16 + m]; b.y  = W2[(kb + 1) * 16 + m];
        acc2 = __builtin_amdgcn_wmma_f32_16x16x4_f32(false, a2, false, b, (short)0, acc2, false, false);
    }

    // D layout: lane (m,kh) holds column j=m of edges base + 8*kh + v
#pragma unroll
    for (int v = 0; v < 8; ++v) {
        const long e2 = base + 8 * kh + v;
        if (e2 < E) {
            const float val = acc2[v];
            eout[e2 * 16 + m] = val;
            const int r2 = row[e2];
            atomicAdd(&node_accum[(size_t)r2 * 16 + m], val);
        }
    }
}

// ---------------------------------------------------------------------------
// Node MLP (scalar; negligible FLOPs). Also folds node sums into graph sums.
//   feat = [x (XW) | node_accum/max(deg,1) (16) | u[batch] (UW)]
// ---------------------------------------------------------------------------
template<int XW, int UW>
__global__ void node_kernel(long N,
                            const float* __restrict__ x, const float* __restrict__ accum,
                            const int* __restrict__ deg,
                            const float* __restrict__ u, const int* __restrict__ batch,
                            const float* __restrict__ W1, const float* __restrict__ B1,
                            const float* __restrict__ W2, const float* __restrict__ B2,
                            float* __restrict__ xout,
                            float* __restrict__ g_n_sum, float* __restrict__ g_e_sum) {
    long n = (long)blockIdx.x * blockDim.x + threadIdx.x;
    if (n >= N) return;
    constexpr int DIN = XW + 16 + UW;
    const int g = batch[n];

    float feat[DIN];
#pragma unroll
    for (int i = 0; i < XW; ++i) feat[i] = x[(size_t)n * XW + i];
    float raw[16];
    const float inv = 1.0f / fmaxf((float)deg[n], 1.0f);
#pragma unroll
    for (int j = 0; j < 16; ++j) {
        raw[j] = accum[(size_t)n * 16 + j];
        feat[XW + j] = raw[j] * inv;
    }
#pragma unroll
    for (int i = 0; i < UW; ++i) feat[XW + 16 + i] = u[(size_t)g * UW + i];

    float h[16];
#pragma unroll
    for (int j = 0; j < 16; ++j) {
        float s = B1[j];
#pragma unroll
        for (int k = 0; k < DIN; ++k) s = fmaf(feat[k], W1[k * 16 + j], s);
        h[j] = fmaxf(s, 0.0f);
    }
#pragma unroll
    for (int j = 0; j < 16; ++j) {
        float s = B2[j];
#pragma unroll
        for (int k = 0; k < 16; ++k) s = fmaf(h[k], W2[k * 16 + j], s);
        xout[(size_t)n * 16 + j] = s;
        atomicAdd(&g_n_sum[g * 16 + j], s);        // mean of x_new over graph
        atomicAdd(&g_e_sum[g * 16 + j], raw[j]);   // sum of ea over graph (via node sums)
    }
}

// ---------------------------------------------------------------------------
// Global MLP:  feat = [u (UW) | mean nodes (16) | mean edges (16)]
// ---------------------------------------------------------------------------
template<int UW, int DOUT>
__global__ void global_kernel(int G, const float* __restrict__ u,
                              const float* __restrict__ g_n_sum, const float* __restrict__ g_n_cnt,
                              const float* __restrict__ g_e_sum, const float* __restrict__ g_e_cnt,
                              const float* __restrict__ W1, const float* __restrict__ B1,
                              const float* __restrict__ W2, const float* __restrict__ B2,
                              float* __restrict__ uout, int do_sigmoid) {
    int g = threadIdx.x;
    if (g >= G) return;
    constexpr int DIN = UW + 32;
    float feat[DIN];
#pragma unroll
    for (int i = 0; i < UW; ++i) feat[i] = u[(size_t)g * UW + i];
    const float invn = 1.0f / fmaxf(g_n_cnt[g], 1.0f);
    const float inve = 1.0f / fmaxf(g_e_cnt[g], 1.0f);
#pragma unroll
    for (int j = 0; j < 16; ++j) {
        feat[UW + j]      = g_n_sum[g * 16 + j] * invn;
        feat[UW + 16 + j] = g_e_sum[g * 16 + j] * inve;
    }
    float h[16];
#pragma unroll
    for (int j = 0; j < 16; ++j) {
        float s = B1[j];
#pragma unroll
        for (int k = 0; k < DIN; ++k) s = fmaf(feat[k], W1[k * 16 + j], s);
        h[j] = fmaxf(s, 0.0f);
    }
#pragma unroll
    for (int j = 0; j < DOUT; ++j) {
        float s = B2[j];
#pragma unroll
        for (int k = 0; k < 16; ++k) s = fmaf(h[k], W2[k * DOUT + j], s);
        if (do_sigmoid) s = 1.0f / (1.0f + expf(-s));
        uout[(size_t)g * DOUT + j] = s;
    }
}

// ---------------------------------------------------------------------------
// Host launcher
// ---------------------------------------------------------------------------
extern "C" void kernel_launch(void* const* d_in, const int* in_sizes, int n_in,
                              void* d_out, int out_size, void* d_ws, size_t ws_size,
                              hipStream_t stream) {
    const long N = in_sizes[0];
    const long E = in_sizes[2];
    const long G = in_sizes[3];

    const float* x0    = (const float*)d_in[0];
    const int*   eidx  = (const int*)d_in[1];
    const int*   row   = eidx;
    const int*   col   = eidx + E;
    const float* ew    = (const float*)d_in[2];
    const float* u0    = (const float*)d_in[3];
    const int*   batch = (const int*)d_in[4];
    // params: e1,n1,g1,e2,n2,g2,e3,n3,g3 ; each {w1,b1,w2,b2}
    const float* P[36];
    for (int i = 0; i < 36; ++i) P[i] = (const float*)d_in[5 + i];

    // Workspace layout
    char* ws = (char*)d_ws;
    size_t o = 0;
    float* ea   = (float*)(ws + o); o += (size_t)E * 16 * 4;
    float* xb   = (float*)(ws + o); o += (size_t)N * 16 * 4;
    float* ub   = (float*)(ws + o); o += (size_t)G * 16 * 4;
    float* nacc = (float*)(ws + o); o += (size_t)N * 16 * 4;
    float* gns  = (float*)(ws + o); o += (size_t)G * 16 * 4;
    float* ges  = (float*)(ws + o); o += (size_t)G * 16 * 4;
    float* gnc  = (float*)(ws + o); o += (size_t)G * 4;
    float* gec  = (float*)(ws + o); o += (size_t)G * 4;
    int*   deg  = (int*)  (ws + o); o += (size_t)N * 4;

    const int TPB = 256;
    const long etiles = (E + 15) / 16;
    const dim3 eGrid((unsigned)((etiles + WPB - 1) / WPB));
    const dim3 nGrid((unsigned)((N + TPB - 1) / TPB));
    const dim3 EGrid((unsigned)((E + TPB - 1) / TPB));

    // ---- one-time (per call) precompute: deg, graph counts ----
    {
        long zc = 2 * G + N;   // gnc, gec, deg are contiguous
        zero_words<<<(unsigned)((zc + TPB - 1) / TPB), TPB, 0, stream>>>((unsigned*)gnc, zc);
        deg_kernel<<<EGrid, TPB, 0, stream>>>(E, row, deg);
        gcnt_kernel<<<nGrid, TPB, 0, stream>>>(N, batch, deg, gnc, gec);
    }

    const long zlayer = N * 16 + 32 * G;   // nacc, gns, ges are contiguous
    const dim3 zGrid((unsigned)((zlayer + TPB - 1) / TPB));

    // ---- layer 1 ----
    zero_words<<<zGrid, TPB, 0, stream>>>((unsigned*)nacc, zlayer);
    edge_kernel<true><<<eGrid, 32 * WPB, 0, stream>>>(E, row, col, batch, x0, ew, u0,
                                                      P[0], P[1], P[2], P[3], ea, nacc);
    node_kernel<1, 1><<<nGrid, TPB, 0, stream>>>(N, x0, nacc, deg, u0, batch,
                                                 P[4], P[5], P[6], P[7], xb, gns, ges);
    global_kernel<1, 16><<<1, 64, 0, stream>>>((int)G, u0, gns, gnc, ges, gec,
                                               P[8], P[9], P[10], P[11], ub, 0);

    // ---- layer 2 ----
    zero_words<<<zGrid, TPB, 0, stream>>>((unsigned*)nacc, zlayer);
    edge_kernel<false><<<eGrid, 32 * WPB, 0, stream>>>(E, row, col, batch, xb, ea, ub,
                                                       P[12], P[13], P[14], P[15], ea, nacc);
    node_kernel<16, 16><<<nGrid, TPB, 0, stream>>>(N, xb, nacc, deg, ub, batch,
                                                   P[16], P[17], P[18], P[19], xb, gns, ges);
    global_kernel<16, 16><<<1, 64, 0, stream>>>((int)G, ub, gns, gnc, ges, gec,
                                                P[20], P[21], P[22], P[23], ub, 0);

    // ---- layer 3 ----
    zero_words<<<zGrid, TPB, 0, stream>>>((unsigned*)nacc, zlayer);
    edge_kernel<false><<<eGrid, 32 * WPB, 0, stream>>>(E, row, col, batch, xb, ea, ub,
                                                       P[24], P[25], P[26], P[27], ea, nacc);
    node_kernel<16, 16><<<nGrid, TPB, 0, stream>>>(N, xb, nacc, deg, ub, batch,
                                                   P[28], P[29], P[30], P[31], xb, gns, ges);
    global_kernel<16, 8><<<1, 64, 0, stream>>>((int)G, ub, gns, gnc, ges, gec,
                                               P[32], P[33], P[34], P[35], (float*)d_out, 1);
}